// Net_55997783605762
// MI455X (gfx1250) — compile-verified
//
#include <hip/hip_runtime.h>
#include <hip/hip_bf16.h>
#include <math.h>

// CDNA5 / gfx1250: wave32, WMMA 16x16x4 f32 path.
typedef __attribute__((ext_vector_type(2))) float v2f;
typedef __attribute__((ext_vector_type(8))) float v8f;

// Per row r:
//   feat = [x0, x1, x2, sin(x2), cos(x2), 1, 1(bias hook), 0]   (K padded 6 -> 8)
//   h[n] = sum_{k<6} feat[k]*W[n][k] + b[n]                      (via 2x WMMA f32 16x16x4)
//   out  = [h0, h1, atan2(softsign(h2), softsign(h3))]
//
// Fast tail: atan2(a/(1+|a|), b/(1+|b|)) needs only ONE v_rcp_f32 because
// softsign is monotonic in |v| and the min/max ratio simplifies to
// (min(|a|,|b|)+|a||b|) / (max(|a|,|b|)+|a||b|).
__global__ __launch_bounds__(256) void net_kernel(const float* __restrict__ xin,
                                                  const float* __restrict__ W,
                                                  const float* __restrict__ bvec,
                                                  float* __restrict__ out,
                                                  int nrows) {
    __shared__ float featLDS[256 * 8];   // 8 KB: 256 rows x 8 feature floats
    __shared__ float hLDS[256 * 4];      // 4 KB: 256 rows x 4 h floats

    const int tid  = threadIdx.x;
    const int lane = tid & 31;
    const int wave = tid >> 5;
    const int row  = blockIdx.x * 256 + tid;
    const int rowL = (row < nrows) ? row : (nrows - 1);   // clamp loads, keep EXEC full

    // ---- 1. Stream x (12 B/row), build feature vector, stage to LDS ----
    const float x0 = xin[3 * rowL + 0];
    const float x1 = xin[3 * rowL + 1];
    const float x2 = xin[3 * rowL + 2];
    float sv, cv;
    __sincosf(x2, &sv, &cv);

    float* f = &featLDS[tid * 8];
    f[0] = x0;   f[1] = x1;   f[2] = x2;   f[3] = sv;
    f[4] = cv;   f[5] = 1.0f; f[6] = 1.0f; f[7] = 0.0f;   // f[6] picks up bias row in B

    __syncthreads();

    // ---- 2. Build B fragments (4x16 f32, lane-striped; uniform across waves) ----
    // Lane L: n = L&15, half = L>>4; VGPR pair holds K = 2*half (+chunk*4) and K+1.
    const int ncol = lane & 15;
    const int half = lane >> 4;
    float w0x = 0.0f, w0y = 0.0f, w1x = 0.0f, w1y = 0.0f;
    if (ncol < 4) {
        w0x = W[ncol * 6 + 2 * half + 0];            // chunk 0: K = {0,1}/{2,3}
        w0y = W[ncol * 6 + 2 * half + 1];
        if (half == 0) {                             // chunk 1: K = {4,5}
            w1x = W[ncol * 6 + 4];
            w1y = W[ncol * 6 + 5];
        } else {                                     // chunk 1: K = {6,7} = bias row, pad
            w1x = bvec[ncol];
            w1y = 0.0f;
        }
    }
    v2f bf0; bf0.x = w0x; bf0.y = w0y;
    v2f bf1; bf1.x = w1x; bf1.y = w1y;

    // ---- 3. Two 16-row M-tiles per wave; 2 WMMAs (K=8) each ----
    const float* wfeat = &featLDS[wave * 32 * 8];
#pragma unroll
    for (int t = 0; t < 2; ++t) {
        const int m0 = t * 16;
        const float* fr = &wfeat[(m0 + ncol) * 8];
        v2f a0; a0.x = fr[2 * half + 0]; a0.y = fr[2 * half + 1];
        v2f a1; a1.x = fr[2 * half + 4]; a1.y = fr[2 * half + 5];

        v8f acc = {};
        // 8 args: (neg_a, A, neg_b, B, c_mod, C, reuse_a, reuse_b)
        acc = __builtin_amdgcn_wmma_f32_16x16x4_f32(false, a0, false, bf0,
                                                    (short)0, acc, false, false);
        acc = __builtin_amdgcn_wmma_f32_16x16x4_f32(false, a1, false, bf1,
                                                    (short)0, acc, false, false);

        // C layout: VGPR r, lane L -> row r + 8*half, col ncol. Keep cols 0..3.
        if (ncol < 4) {
            float* hb = &hLDS[(wave * 32 + m0 + half * 8) * 4 + ncol];
#pragma unroll
            for (int r = 0; r < 8; ++r) hb[r * 4] = acc[r];
        }
    }

    __syncthreads();

    // ---- 4. Fast softsign-atan2 tail + 12 B/row streaming store ----
    if (row < nrows) {
        const float* h = &hLDS[tid * 4];
        const float h2 = h[2], h3 = h[3];

        const float a  = fabsf(h2), b = fabsf(h3);
        const float ab = a * b;
        const float mn = fminf(a, b), mx = fmaxf(a, b);
        // t = min(|s2|,|s3|)/max(|s2|,|s3|), one v_rcp_f32 total
        const float tq = (mn + ab) * __builtin_amdgcn_rcpf(mx + ab);
        const float sq = tq * tq;
        // minimax atan(t) on [0,1]
        float r = fmaf(sq, -0.01172120f, 0.05265332f);
        r = fmaf(sq, r, -0.11643287f);
        r = fmaf(sq, r, 0.19354346f);
        r = fmaf(sq, r, -0.33262347f);
        r = fmaf(sq, r, 0.99997726f);
        r = r * tq;
        if (a > b)       r = 1.57079637f - r;   // |y| > |x|: reflect about pi/2
        if (h3 < 0.0f)   r = 3.14159274f - r;   // x < 0: reflect about pi
        r = copysignf(r, h2);                   // sign of y

        out[3 * row + 0] = h[0];
        out[3 * row + 1] = h[1];
        out[3 * row + 2] = r;
    }
}

extern "C" void kernel_launch(void* const* d_in, const int* in_sizes, int n_in,
                              void* d_out, int out_size, void* d_ws, size_t ws_size,
                              hipStream_t stream) {
    const float* x = (const float*)d_in[0];   // (B, 3) f32
    const float* W = (const float*)d_in[1];   // (4, 6) f32
    const float* b = (const float*)d_in[2];   // (4,)  f32
    float* out = (float*)d_out;               // (B, 3) f32

    const int nrows = in_sizes[0] / 3;
    const int blocks = (nrows + 255) / 256;
    net_kernel<<<blocks, 256, 0, stream>>>(x, W, b, out, nrows);
}